// DeltaConvEncoder_62088047231001
// MI455X (gfx1250) — compile-verified
//
#include <hip/hip_runtime.h>

// DeltaConvEncoder for MI455X (gfx1250, wave32).
//
// Memory-bound problem: ~0.3 GFLOP vs ~68MB traffic -> ~3us at 23.3 TB/s.
//  pass 1: stats_kernel  -- LDS-tiled read of x; per-block partial sums for
//          bn1 (scalar over delta) and per-t conv bn2 (conv bias provably
//          cancels inside training-mode BN). Fixed-order reductions only
//          (no float atomics) so results are bit-deterministic per call.
//  pass 2: finalize_kernel -- fixed-order reduction of block partials, folds
//          BN + linear-proj into per-t scale/shift coefficients.
//  pass 3: main_kernel   -- conv via V_WMMA_F32_16X16X4_F32 (A = 16x4 weight
//          matrix, K=3 padded to 4; B = 4x16 input window from LDS), half-wave
//          shfl_xor (ds_bpermute) so each lane holds all 16 t-values of its
//          column, dual 16-step LIF scan in registers, coalesced 128B stores.

typedef float v2f __attribute__((ext_vector_type(2)));
typedef float v8f __attribute__((ext_vector_type(8)));

constexpr int B = 32, L = 512, C = 64, T = 16;
constexpr int CHUNK = 32;              // l-positions per tile
constexpr int ROWS = CHUNK + 2;        // halo rows (l0-1 .. l0+32)
constexpr int LS = 65;                 // LDS row stride (bank-conflict pad)
constexpr int NCHUNK = L / CHUNK;      // 16
constexpr int NTILES = B * NCHUNK;     // 512
constexpr int STILES = 4;              // tiles per stats block
constexpr int SBLOCKS = NTILES / STILES; // 128 stats blocks
constexpr int NACC = 34;               // 2 bn1 + 16 sum + 16 sumsq
constexpr float EPS = 1e-5f;

__device__ __forceinline__ float wave_sum(float v) {
  #pragma unroll
  for (int m = 16; m >= 1; m >>= 1) v += __shfl_xor(v, m, 32);
  return v;
}

__global__ __launch_bounds__(256) void stats_kernel(
    const float* __restrict__ x, const float* __restrict__ cw,
    float* __restrict__ part) {
  __shared__ float tile[ROWS * LS];
  __shared__ float red[8 * NACC];
  const int tid = threadIdx.x;

  float w0[T], w1[T], w2[T];
  #pragma unroll
  for (int t = 0; t < T; ++t) {
    w0[t] = cw[t * 3 + 0]; w1[t] = cw[t * 3 + 1]; w2[t] = cw[t * 3 + 2];
  }

  float s1 = 0.f, s1q = 0.f;
  float sy[T], syq[T];
  #pragma unroll
  for (int t = 0; t < T; ++t) { sy[t] = 0.f; syq[t] = 0.f; }

  for (int tt = 0; tt < STILES; ++tt) {
    const int tileid = blockIdx.x * STILES + tt;
    const int b  = tileid / NCHUNK;
    const int l0 = (tileid % NCHUNK) * CHUNK;

    __syncthreads();   // protect tile vs previous iteration's readers
    for (int i = tid; i < ROWS * C; i += 256) {
      int r = i >> 6, c = i & 63;
      int l = l0 - 1 + r;
      tile[r * LS + c] = (l >= 0 && l < L) ? x[(b * L + l) * C + c] : 0.0f;
    }
    __syncthreads();

    #pragma unroll
    for (int i = 0; i < 8; ++i) {
      int idx = tid + i * 256;
      int p = idx >> 6, c = idx & 63;          // p in [0,32), c in [0,64)
      float xm = tile[(p    ) * LS + c];       // position p-1
      float x0 = tile[(p + 1) * LS + c];       // position p
      float xp = tile[(p + 2) * LS + c];       // position p+1
      float d = (l0 + p == 0) ? 0.0f : (x0 - xm);
      s1 += d; s1q += d * d;
      #pragma unroll
      for (int t = 0; t < T; ++t) {
        float y = w0[t] * xm + w1[t] * x0 + w2[t] * xp;  // raw conv (no bias)
        sy[t] += y; syq[t] += y * y;
      }
    }
  }

  // Wave-level reduction (fixed shuffle tree), then fixed-order block sum.
  s1 = wave_sum(s1); s1q = wave_sum(s1q);
  #pragma unroll
  for (int t = 0; t < T; ++t) { sy[t] = wave_sum(sy[t]); syq[t] = wave_sum(syq[t]); }

  const int lane = tid & 31, wv = tid >> 5;
  if (lane == 0) {
    red[wv * NACC + 0] = s1;
    red[wv * NACC + 1] = s1q;
    #pragma unroll
    for (int t = 0; t < T; ++t) {
      red[wv * NACC + 2 + t]  = sy[t];
      red[wv * NACC + 18 + t] = syq[t];
    }
  }
  __syncthreads();
  if (tid < NACC) {
    float s = 0.f;
    #pragma unroll
    for (int w = 0; w < 8; ++w) s += red[w * NACC + tid];
    part[blockIdx.x * NACC + tid] = s;
  }
}

__global__ __launch_bounds__(64) void finalize_kernel(
    const float* __restrict__ part,
    const float* __restrict__ g1, const float* __restrict__ b1,
    const float* __restrict__ pw, const float* __restrict__ pb,
    const float* __restrict__ g2, const float* __restrict__ b2,
    float* __restrict__ coef) {
  __shared__ float acc[NACC];
  const int tid = threadIdx.x;
  if (tid < NACC) {                       // fixed-order deterministic sum
    float s = 0.f;
    for (int i = 0; i < SBLOCKS; ++i) s += part[i * NACC + tid];
    acc[tid] = s;
  }
  __syncthreads();
  if (tid >= T) return;
  const int t = tid;
  const float invN = 1.0f / (float)(B * L * C);
  // bn1 (scalar over whole delta tensor)
  float mu1  = acc[0] * invN;
  float var1 = acc[1] * invN - mu1 * mu1;
  float sc1  = g1[0] * rsqrtf(var1 + EPS);
  float sh1  = b1[0] - mu1 * sc1;
  // bn2 per channel t (conv bias cancels: BN subtracts its own mean)
  float mu2  = acc[2 + t]  * invN;
  float var2 = acc[18 + t] * invN - mu2 * mu2;
  float sc2  = g2[t] * rsqrtf(var2 + EPS);
  float sh2  = b2[t] - mu2 * sc2;
  coef[t]      = sc2;                     // conv scale
  coef[16 + t] = sh2;                     // conv shift
  coef[32 + t] = sc1 * pw[t];             // delta scale (BN folded into proj)
  coef[48 + t] = sh1 * pw[t] + pb[t];     // delta shift
}

__global__ __launch_bounds__(256) void main_kernel(
    const float* __restrict__ x, const float* __restrict__ cw,
    const float* __restrict__ coef, float* __restrict__ out) {
  __shared__ float tile[ROWS * LS];
  __shared__ float cf[64];
  const int b  = blockIdx.x / NCHUNK;
  const int l0 = (blockIdx.x % NCHUNK) * CHUNK;
  const int tid = threadIdx.x;

  if (tid < 64) cf[tid] = coef[tid];
  for (int i = tid; i < ROWS * C; i += 256) {
    int r = i >> 6, c = i & 63;
    int l = l0 - 1 + r;
    tile[r * LS + c] = (l >= 0 && l < L) ? x[(b * L + l) * C + c] : 0.0f;
  }
  __syncthreads();

  const int lane = tid & 31;
  const int wv   = tid >> 5;        // wave in block: 0..7
  const int m    = lane & 15;
  const bool lo  = lane < 16;

  // A operand: 16x4 f32 weight matrix (K=3 padded to 4).
  // Layout: VGPR0 = K0 (lanes 0-15) | K2 (lanes 16-31); VGPR1 = K1 | K3(=0).
  v2f a;
  a.x = lo ? cw[m * 3 + 0] : cw[m * 3 + 2];
  a.y = lo ? cw[m * 3 + 1] : 0.0f;

  // B row offset for this lane: K0 needs x(p-1), K2 needs x(p+1).
  const int offx = lo ? (m - 1) : (m + 1);

  for (int ci = 0; ci < 8; ++ci) {
    const int c = wv + ci * 8;
    const float* col = &tile[c];          // col[row * LS], row = position + 1

    // B operands: tile0 covers positions 0..15, tile1 covers 16..31.
    v2f b0, b1;
    b0.x = col[(0  + offx + 1) * LS];
    b0.y = lo ? col[(0  + m + 1) * LS] : 0.0f;
    b1.x = col[(16 + offx + 1) * LS];
    b1.y = lo ? col[(16 + m + 1) * LS] : 0.0f;

    v8f cz = {};
    v8f d0 = __builtin_amdgcn_wmma_f32_16x16x4_f32(
        false, a, false, b0, (short)0, cz, false, false);
    v8f d1 = __builtin_amdgcn_wmma_f32_16x16x4_f32(
        false, a, false, b1, (short)0, cz, false, false);

    // Half-wave exchange: lane ends holding all 16 t-values of its own column
    // (lane n -> position l0+n; lane 16+n -> position l0+16+n).
    float y16[16];
    #pragma unroll
    for (int v = 0; v < 8; ++v) {
      float e0 = d0[v], e1 = d1[v];
      float o0 = __shfl_xor(e0, 16, 32);
      float o1 = __shfl_xor(e1, 16, 32);
      y16[v]     = lo ? e0 : o1;   // t = v
      y16[v + 8] = lo ? o0 : e1;   // t = v + 8
    }

    // Delta branch for own column.
    float x0 = col[(lane + 1) * LS];
    float xm = col[(lane    ) * LS];
    float dl = (l0 + lane == 0) ? 0.0f : (x0 - xm);

    // Dual LIF scan over T (beta=0.5, thresh=1, reset-by-subtract).
    float memA = 0.0f, memB = 0.0f;
    const int l = l0 + lane;
    const size_t obase = (size_t)b * T * C * L + (size_t)c * L + l;
    #pragma unroll
    for (int t = 0; t < T; ++t) {
      float xa = dl * cf[32 + t] + cf[48 + t];
      float xb = y16[t] * cf[t] + cf[16 + t];
      float rA = memA > 1.0f ? 1.0f : 0.0f;
      memA = 0.5f * memA + xa - rA;
      float sA = memA > 1.0f ? 1.0f : 0.0f;
      float rB = memB > 1.0f ? 1.0f : 0.0f;
      memB = 0.5f * memB + xb - rB;
      float sB = memB > 1.0f ? 1.0f : 0.0f;
      out[obase + (size_t)t * C * L] = sA + sB;
    }
  }
}

extern "C" void kernel_launch(void* const* d_in, const int* in_sizes, int n_in,
                              void* d_out, int out_size, void* d_ws, size_t ws_size,
                              hipStream_t stream) {
  (void)in_sizes; (void)n_in; (void)out_size; (void)ws_size;
  const float* x  = (const float*)d_in[0];
  const float* g1 = (const float*)d_in[1];
  const float* b1 = (const float*)d_in[2];
  const float* pw = (const float*)d_in[3];
  const float* pb = (const float*)d_in[4];
  const float* cw = (const float*)d_in[5];
  // d_in[6] = conv_b: provably a no-op under training-mode BatchNorm.
  const float* g2 = (const float*)d_in[7];
  const float* b2 = (const float*)d_in[8];

  float* wsf  = (float*)d_ws;
  float* part = wsf;                       // SBLOCKS * NACC = 4352 floats
  float* coef = wsf + SBLOCKS * NACC;      // 64 floats
  float* o    = (float*)d_out;

  hipLaunchKernelGGL(stats_kernel, dim3(SBLOCKS), dim3(256), 0, stream, x, cw, part);
  hipLaunchKernelGGL(finalize_kernel, dim3(1), dim3(64), 0, stream,
                     part, g1, b1, pw, pb, g2, b2, coef);
  hipLaunchKernelGGL(main_kernel, dim3(NTILES), dim3(256), 0, stream, x, cw, coef, o);
}